// DGCNN_encoder_77318001263008
// MI455X (gfx1250) — compile-verified
//
#include <hip/hip_runtime.h>

typedef __attribute__((ext_vector_type(2))) float v2f;
typedef __attribute__((ext_vector_type(8))) float v8f;

#define BATCH 8
#define NPTS  1024
#define KNN_K 20
#define PKN   (NPTS * KNN_K)   // 20480 positions per batch for edge tensors
#define EPSBN 1e-5f

__device__ __forceinline__ float lrelu_f(float y) { return y < 0.f ? 0.2f * y : y; }

__device__ __forceinline__ v8f wmma4(v2f a, v2f b, v8f c) {
    return __builtin_amdgcn_wmma_f32_16x16x4_f32(false, a, false, b, (short)0, c,
                                                 false, false);
}

// ---------------------------------------------------------------------------
// Input x (B, N, 3) row-major -> xt (B, 4, N) with channel 3 zero-padded.
// ---------------------------------------------------------------------------
__global__ void k_transpose_in4(const float* __restrict__ x, float* __restrict__ xt) {
    int t = blockIdx.x * blockDim.x + threadIdx.x;
    if (t >= BATCH * 4 * NPTS) return;
    int b = t / (4 * NPTS);
    int r = t % (4 * NPTS);
    int c = r / NPTS;
    int n = r % NPTS;
    xt[t] = (c < 3) ? x[((size_t)b * NPTS + n) * 3 + c] : 0.f;
}

// ---------------------------------------------------------------------------
// Repack (64, 6) edge weights -> (64, 8): cols [w0 w1 w2 0 | w3 w4 w5 0]
// matching the 4-channel zero-padded feature layout.
// ---------------------------------------------------------------------------
__global__ void k_pad_w6to8(const float* __restrict__ w6, float* __restrict__ w8) {
    int t = blockIdx.x * blockDim.x + threadIdx.x;
    if (t >= 64 * 8) return;
    int o = t / 8, c = t % 8;
    float v = 0.f;
    if (c < 3)               v = w6[o * 6 + c];
    else if (c >= 4 && c < 7) v = w6[o * 6 + (c - 1)];
    w8[t] = v;
}

// ---------------------------------------------------------------------------
// Per-point squared norms: nrm[b,n] = sum_c feat[b,c,n]^2  (zero pad harmless)
// ---------------------------------------------------------------------------
template <int C>
__global__ void k_norms(const float* __restrict__ feat, int bstride,
                        float* __restrict__ nrm) {
    int t = blockIdx.x * blockDim.x + threadIdx.x;
    if (t >= BATCH * NPTS) return;
    int b = t / NPTS, n = t % NPTS;
    const float* X = feat + (size_t)b * bstride;
    float s = 0.f;
    #pragma unroll
    for (int c = 0; c < C; ++c) {
        float v = X[c * NPTS + n];
        s += v * v;
    }
    nrm[t] = s;
}

// ---------------------------------------------------------------------------
// Gram matrix G[b,i,j] = sum_c X[b,c,i]*X[b,c,j] via V_WMMA_F32_16X16X4_F32.
// Each wave: 16 rows x 64 cols (4 tiles), A reused across 4 WMMAs per chunk.
// Block = 4 waves (256 cols). Grid: (N/256, N/16, B). C must be mult of 4.
// A lane layout (ISA 16x4 f32): lanes 0-15 hold K=kk,kk+1; lanes 16-31 kk+2,kk+3.
// ---------------------------------------------------------------------------
template <int C>
__global__ void k_gram_wmma(const float* __restrict__ feat, int bstride,
                            float* __restrict__ gram) {
    int lane = threadIdx.x & 31;
    int wv   = threadIdx.x >> 5;
    int half = lane >> 4;
    int m    = lane & 15;
    int j0   = (blockIdx.x * 4 + wv) * 64;
    int i0   = blockIdx.y * 16;
    int b    = blockIdx.z;
    const float* X = feat + (size_t)b * bstride;

    v8f acc0 = {0.f,0.f,0.f,0.f,0.f,0.f,0.f,0.f};
    v8f acc1 = acc0, acc2 = acc0, acc3 = acc0;

    for (int kk = 0; kk < C; kk += 4) {
        int ka = kk + 2 * half;
        const float* Xa = X + (size_t)ka * NPTS;
        const float* Xb = Xa + NPTS;
        v2f a, b0, b1, b2, b3;
        a[0]  = Xa[i0 + m];       a[1]  = Xb[i0 + m];
        b0[0] = Xa[j0 + m];       b0[1] = Xb[j0 + m];
        b1[0] = Xa[j0 + 16 + m];  b1[1] = Xb[j0 + 16 + m];
        b2[0] = Xa[j0 + 32 + m];  b2[1] = Xb[j0 + 32 + m];
        b3[0] = Xa[j0 + 48 + m];  b3[1] = Xb[j0 + 48 + m];
        acc0 = wmma4(a, b0, acc0);
        acc1 = wmma4(a, b1, acc1);
        acc2 = wmma4(a, b2, acc2);
        acc3 = wmma4(a, b3, acc3);
    }
    float* G = gram + (size_t)b * NPTS * NPTS;
    #pragma unroll
    for (int r = 0; r < 8; ++r) {
        size_t row = (size_t)(i0 + r + 8 * half) * NPTS;
        G[row + j0 + m]      = acc0[r];
        G[row + j0 + 16 + m] = acc1[r];
        G[row + j0 + 32 + m] = acc2[r];
        G[row + j0 + 48 + m] = acc3[r];
    }
}

// ---------------------------------------------------------------------------
// Top-K (largest pd = 2*dot - |xi|^2 - |xj|^2) brute-force per (b,i) thread.
// ---------------------------------------------------------------------------
__global__ void k_topk(const float* __restrict__ gram, const float* __restrict__ nrm,
                       int* __restrict__ idx) {
    int t = blockIdx.x * blockDim.x + threadIdx.x;
    if (t >= BATCH * NPTS) return;
    int b = t / NPTS, i = t % NPTS;
    const float* G  = gram + (size_t)b * NPTS * NPTS + (size_t)i * NPTS;
    const float* Nr = nrm + b * NPTS;
    float ni = Nr[i];

    float best[KNN_K];
    int   bidx[KNN_K];
    #pragma unroll
    for (int q = 0; q < KNN_K; ++q) { best[q] = -3.4e38f; bidx[q] = 0; }

    for (int j = 0; j < NPTS; ++j) {
        float pd = 2.f * G[j] - ni - Nr[j];
        if (pd > best[KNN_K - 1]) {
            int pos = KNN_K - 1;
            while (pos > 0 && best[pos - 1] < pd) {
                best[pos] = best[pos - 1];
                bidx[pos] = bidx[pos - 1];
                --pos;
            }
            best[pos] = pd;
            bidx[pos] = j;
        }
    }
    int* out = idx + (size_t)b * PKN + (size_t)i * KNN_K;
    #pragma unroll
    for (int q = 0; q < KNN_K; ++q) out[q] = bidx[q];
}

// ---------------------------------------------------------------------------
// Branchless fused graph-feature value: both loads from valid addresses,
// ternary selects VALUES only (v_cndmask, no EXEC manipulation).
//   c <  C : x[c, j] - x[c, i]     (difference to neighbor)
//   c >= C : x[c-C, i]             (center copy)
// ---------------------------------------------------------------------------
template <int C>
__device__ __forceinline__ float edge_feat(const float* __restrict__ X,
                                           int c, int i, int j) {
    bool diff = (c < C);
    int ch = diff ? c : (c - C);
    float vi = X[ch * NPTS + i];
    float vj = X[ch * NPTS + j];
    return diff ? (vj - vi) : vi;
}

// ---------------------------------------------------------------------------
// Edge conv: out[b,o,p] = sum_{c'<2C} W[o,c'] * f(c', p), graph gather fused.
// Each wave: 16 out-channels x 64 positions (4 tiles). Grid: (PKN/256, Cout/16, B).
// W has 2C columns (pre-padded to a multiple of 4).
// ---------------------------------------------------------------------------
template <int C>
__global__ void k_edgeconv_wmma(const float* __restrict__ feat, int bstride,
                                const int* __restrict__ idx,
                                const float* __restrict__ w, int Cout,
                                float* __restrict__ out) {
    int lane = threadIdx.x & 31;
    int wv   = threadIdx.x >> 5;
    int half = lane >> 4;
    int m    = lane & 15;
    int p0   = (blockIdx.x * 4 + wv) * 64;
    int o0   = blockIdx.y * 16;
    int b    = blockIdx.z;

    const float* X  = feat + (size_t)b * bstride;
    const float* Wr = w + (size_t)(o0 + m) * (2 * C);
    const int*   Ib = idx + (size_t)b * PKN;

    int pi[4], pj[4];
    #pragma unroll
    for (int t = 0; t < 4; ++t) {
        int p = p0 + 16 * t + m;
        pi[t] = p / KNN_K;
        pj[t] = Ib[p];
    }

    v8f acc0 = {0.f,0.f,0.f,0.f,0.f,0.f,0.f,0.f};
    v8f acc1 = acc0, acc2 = acc0, acc3 = acc0;

    for (int kk = 0; kk < 2 * C; kk += 4) {
        int ka = kk + 2 * half;
        v2f a, b0, b1, b2, b3;
        a[0]  = Wr[ka];
        a[1]  = Wr[ka + 1];
        b0[0] = edge_feat<C>(X, ka,     pi[0], pj[0]);
        b0[1] = edge_feat<C>(X, ka + 1, pi[0], pj[0]);
        b1[0] = edge_feat<C>(X, ka,     pi[1], pj[1]);
        b1[1] = edge_feat<C>(X, ka + 1, pi[1], pj[1]);
        b2[0] = edge_feat<C>(X, ka,     pi[2], pj[2]);
        b2[1] = edge_feat<C>(X, ka + 1, pi[2], pj[2]);
        b3[0] = edge_feat<C>(X, ka,     pi[3], pj[3]);
        b3[1] = edge_feat<C>(X, ka + 1, pi[3], pj[3]);
        acc0 = wmma4(a, b0, acc0);
        acc1 = wmma4(a, b1, acc1);
        acc2 = wmma4(a, b2, acc2);
        acc3 = wmma4(a, b3, acc3);
    }
    float* O = out + (size_t)b * Cout * PKN;
    #pragma unroll
    for (int r = 0; r < 8; ++r) {
        size_t row = (size_t)(o0 + r + 8 * half) * PKN;
        O[row + p0 + m]      = acc0[r];
        O[row + p0 + 16 + m] = acc1[r];
        O[row + p0 + 32 + m] = acc2[r];
        O[row + p0 + 48 + m] = acc3[r];
    }
}

// ---------------------------------------------------------------------------
// Plain 1x1 conv over contiguous (B, CIN, P). Grid: (P/256, Cout/16, B).
// ---------------------------------------------------------------------------
template <int CIN>
__global__ void k_conv_wmma(const float* __restrict__ in, int P,
                            const float* __restrict__ w, int Cout,
                            float* __restrict__ out) {
    int lane = threadIdx.x & 31;
    int wv   = threadIdx.x >> 5;
    int half = lane >> 4;
    int m    = lane & 15;
    int p0   = (blockIdx.x * 4 + wv) * 64;
    int o0   = blockIdx.y * 16;
    int b    = blockIdx.z;

    const float* I  = in + (size_t)b * CIN * P;
    const float* Wr = w + (size_t)(o0 + m) * CIN;

    v8f acc0 = {0.f,0.f,0.f,0.f,0.f,0.f,0.f,0.f};
    v8f acc1 = acc0, acc2 = acc0, acc3 = acc0;

    for (int kk = 0; kk < CIN; kk += 4) {
        int ka = kk + 2 * half;
        const float* Ia = I + (size_t)ka * P;
        const float* Ib2 = Ia + P;
        v2f a, b0, b1, b2, b3;
        a[0]  = Wr[ka];           a[1]  = Wr[ka + 1];
        b0[0] = Ia[p0 + m];       b0[1] = Ib2[p0 + m];
        b1[0] = Ia[p0 + 16 + m];  b1[1] = Ib2[p0 + 16 + m];
        b2[0] = Ia[p0 + 32 + m];  b2[1] = Ib2[p0 + 32 + m];
        b3[0] = Ia[p0 + 48 + m];  b3[1] = Ib2[p0 + 48 + m];
        acc0 = wmma4(a, b0, acc0);
        acc1 = wmma4(a, b1, acc1);
        acc2 = wmma4(a, b2, acc2);
        acc3 = wmma4(a, b3, acc3);
    }
    float* O = out + (size_t)b * Cout * P;
    #pragma unroll
    for (int r = 0; r < 8; ++r) {
        size_t row = (size_t)(o0 + r + 8 * half) * P;
        O[row + p0 + m]      = acc0[r];
        O[row + p0 + 16 + m] = acc1[r];
        O[row + p0 + 32 + m] = acc2[r];
        O[row + p0 + 48 + m] = acc3[r];
    }
}

// ---------------------------------------------------------------------------
// Per-channel sum / sum-of-squares over (B, C, P). One block per channel.
// ---------------------------------------------------------------------------
__global__ void k_chan_stats(const float* __restrict__ e, int C, int P,
                             float* __restrict__ stats) {
    int c = blockIdx.x;
    __shared__ float s1[256];
    __shared__ float s2[256];
    float a = 0.f, q = 0.f;
    for (int b = 0; b < BATCH; ++b) {
        const float* row = e + ((size_t)b * C + c) * (size_t)P;
        for (int p = threadIdx.x; p < P; p += 256) {
            float v = row[p];
            a += v;
            q += v * v;
        }
    }
    s1[threadIdx.x] = a;
    s2[threadIdx.x] = q;
    __syncthreads();
    for (int s = 128; s > 0; s >>= 1) {
        if (threadIdx.x < s) {
            s1[threadIdx.x] += s1[threadIdx.x + s];
            s2[threadIdx.x] += s2[threadIdx.x + s];
        }
        __syncthreads();
    }
    if (threadIdx.x == 0) {
        stats[2 * c]     = s1[0];
        stats[2 * c + 1] = s2[0];
    }
}

// ---------------------------------------------------------------------------
// In-place batch-norm (batch statistics) + leaky ReLU over (B, C, P).
// ---------------------------------------------------------------------------
__global__ void k_bn_lrelu(float* __restrict__ e, int C, int P,
                           const float* __restrict__ stats,
                           const float* __restrict__ g,
                           const float* __restrict__ bb) {
    size_t t = (size_t)blockIdx.x * 256 + threadIdx.x;
    size_t total = (size_t)BATCH * C * P;
    if (t >= total) return;
    int c = (int)((t / (size_t)P) % (size_t)C);
    float cnt  = (float)BATCH * (float)P;
    float mean = stats[2 * c] / cnt;
    float var  = stats[2 * c + 1] / cnt - mean * mean;
    float inv  = rsqrtf(var + EPSBN);
    float y = g[c] * (e[t] - mean) * inv + bb[c];
    e[t] = lrelu_f(y);
}

// ---------------------------------------------------------------------------
// Max over the K (neighbor) axis: (B,C,N,K) -> dst[b*dst_bstride + c*N + n]
// ---------------------------------------------------------------------------
__global__ void k_maxpool_k(const float* __restrict__ e, int C,
                            float* __restrict__ dst, int dst_bstride) {
    int t = blockIdx.x * blockDim.x + threadIdx.x;
    if (t >= BATCH * C * NPTS) return;
    int b = t / (C * NPTS);
    int r = t % (C * NPTS);
    int c = r / NPTS;
    int n = r % NPTS;
    const float* p_ = e + ((size_t)b * C + c) * (size_t)PKN + (size_t)n * KNN_K;
    float mx = p_[0];
    #pragma unroll
    for (int kk = 1; kk < KNN_K; ++kk) mx = fmaxf(mx, p_[kk]);
    dst[(size_t)b * dst_bstride + (size_t)c * NPTS + n] = mx;
}

// ---------------------------------------------------------------------------
// Max over the N axis: (B,C,N) -> out (B,C)
// ---------------------------------------------------------------------------
__global__ void k_maxpool_n(const float* __restrict__ e, int C,
                            float* __restrict__ out) {
    int t = blockIdx.x * blockDim.x + threadIdx.x;
    if (t >= BATCH * C) return;
    int b = t / C, c = t % C;
    const float* p_ = e + ((size_t)b * C + c) * (size_t)NPTS;
    float mx = p_[0];
    for (int n = 1; n < NPTS; ++n) mx = fmaxf(mx, p_[n]);
    out[t] = mx;
}

// ---------------------------------------------------------------------------
// Small fully-connected: out[b,o] = dot(in[b,:], w[o,:]) (+ bias)
// ---------------------------------------------------------------------------
__global__ void k_fc(const float* __restrict__ in, int Cin,
                     const float* __restrict__ w, const float* __restrict__ bias,
                     int Cout, float* __restrict__ out) {
    int t = blockIdx.x * blockDim.x + threadIdx.x;
    if (t >= BATCH * Cout) return;
    int b = t / Cout, o = t % Cout;
    const float* I  = in + (size_t)b * Cin;
    const float* Wr = w + (size_t)o * Cin;
    float s = 0.f;
    for (int c = 0; c < Cin; ++c) s += I[c] * Wr[c];
    if (bias) s += bias[o];
    out[t] = s;
}

// ---------------------------------------------------------------------------
// Batch-norm (over batch of 8) + leaky ReLU for (B, C) activations, in place.
// ---------------------------------------------------------------------------
__global__ void k_bn_vec_lrelu(float* __restrict__ f, int C,
                               const float* __restrict__ g,
                               const float* __restrict__ bb) {
    int c = blockIdx.x * blockDim.x + threadIdx.x;
    if (c >= C) return;
    float s = 0.f, q = 0.f;
    for (int b = 0; b < BATCH; ++b) {
        float v = f[(size_t)b * C + c];
        s += v;
        q += v * v;
    }
    float mean = s / (float)BATCH;
    float var  = q / (float)BATCH - mean * mean;
    float inv  = rsqrtf(var + EPSBN);
    for (int b = 0; b < BATCH; ++b) {
        float y = g[c] * (f[(size_t)b * C + c] - mean) * inv + bb[c];
        f[(size_t)b * C + c] = lrelu_f(y);
    }
}

// ---------------------------------------------------------------------------
// fc3 (256 -> 9) + bias + identity(3x3) add -> tmat (B, 9)
// ---------------------------------------------------------------------------
__global__ void k_fc3_eye(const float* __restrict__ in, const float* __restrict__ w,
                          const float* __restrict__ bias, float* __restrict__ tmat) {
    int t = blockIdx.x * blockDim.x + threadIdx.x;
    if (t >= BATCH * 9) return;
    int b = t / 9, o = t % 9;
    float s = bias[o] + ((o == 0 || o == 4 || o == 8) ? 1.f : 0.f);
    const float* I  = in + (size_t)b * 256;
    const float* Wr = w + (size_t)o * 256;
    for (int c = 0; c < 256; ++c) s += I[c] * Wr[c];
    tmat[t] = s;
}

// ---------------------------------------------------------------------------
// Apply learned 3x3 transform: xt2 (B,4,N), ch 3 zero-padded.
// xt2[b,d,n] = sum_j tmat[b,d,j] * xt[b,j,n]
// ---------------------------------------------------------------------------
__global__ void k_apply_transform(const float* __restrict__ xt,
                                  const float* __restrict__ tmat,
                                  float* __restrict__ xt2) {
    int t = blockIdx.x * blockDim.x + threadIdx.x;
    if (t >= BATCH * 4 * NPTS) return;
    int b = t / (4 * NPTS);
    int r = t % (4 * NPTS);
    int d = r / NPTS;
    int n = r % NPTS;
    float s = 0.f;
    if (d < 3) {
        #pragma unroll
        for (int j = 0; j < 3; ++j)
            s += tmat[b * 9 + d * 3 + j] * xt[(size_t)b * 4 * NPTS + (size_t)j * NPTS + n];
    }
    xt2[t] = s;
}

// ===========================================================================
// Host-side helpers (template dispatch)
// ===========================================================================
template <int C>
static void run_knn(const float* feat, int bstride, float* nrm, float* gram,
                    int* idx, hipStream_t stream) {
    k_norms<C><<<dim3((BATCH * NPTS + 255) / 256), 256, 0, stream>>>(feat, bstride, nrm);
    k_gram_wmma<C><<<dim3(NPTS / 256, NPTS / 16, BATCH), 128, 0, stream>>>(feat, bstride, gram);
    k_topk<<<dim3((BATCH * NPTS + 127) / 128), 128, 0, stream>>>(gram, nrm, idx);
}

static void run_bn4d(float* e, int C, int P, const float* g, const float* bb,
                     float* stats, hipStream_t stream) {
    k_chan_stats<<<dim3(C), 256, 0, stream>>>(e, C, P, stats);
    size_t total = (size_t)BATCH * C * P;
    k_bn_lrelu<<<dim3((unsigned)((total + 255) / 256)), 256, 0, stream>>>(e, C, P, stats, g, bb);
}

template <int C>
static void run_edgeconv(const float* feat, int bstride, const float* w, int Cout,
                         const float* g, const float* bb, float* dst, int dst_bstride,
                         float* nrm, float* gram, int* idx, float* ebuf, float* stats,
                         hipStream_t stream) {
    run_knn<C>(feat, bstride, nrm, gram, idx, stream);
    k_edgeconv_wmma<C><<<dim3(PKN / 256, Cout / 16, BATCH), 128, 0, stream>>>(
        feat, bstride, idx, w, Cout, ebuf);
    run_bn4d(ebuf, Cout, PKN, g, bb, stats, stream);
    k_maxpool_k<<<dim3((BATCH * Cout * NPTS + 255) / 256), 256, 0, stream>>>(
        ebuf, Cout, dst, dst_bstride);
}

// ===========================================================================
extern "C" void kernel_launch(void* const* d_in, const int* in_sizes, int n_in,
                              void* d_out, int out_size, void* d_ws, size_t ws_size,
                              hipStream_t stream) {
    (void)in_sizes; (void)n_in; (void)out_size; (void)ws_size;
    const float* x_in   = (const float*)d_in[0];
    const float* Wt1    = (const float*)d_in[1];   // (64, 6)
    const float* tbn1g  = (const float*)d_in[2];
    const float* tbn1b  = (const float*)d_in[3];
    const float* Wt2    = (const float*)d_in[4];   // (128, 64)
    const float* tbn2g  = (const float*)d_in[5];
    const float* tbn2b  = (const float*)d_in[6];
    const float* Wt3    = (const float*)d_in[7];   // (1024, 128)
    const float* tbn3g  = (const float*)d_in[8];
    const float* tbn3b  = (const float*)d_in[9];
    const float* Wf1    = (const float*)d_in[10];  // (512, 1024)
    const float* bnf1g  = (const float*)d_in[11];
    const float* bnf1b  = (const float*)d_in[12];
    const float* Wf2    = (const float*)d_in[13];  // (256, 512)
    const float* bf2    = (const float*)d_in[14];
    const float* bnf2g  = (const float*)d_in[15];
    const float* bnf2b  = (const float*)d_in[16];
    const float* Wf3    = (const float*)d_in[17];  // (9, 256)
    const float* bf3    = (const float*)d_in[18];
    const float* Wc1    = (const float*)d_in[19];  // (64, 6)
    const float* bn1g   = (const float*)d_in[20];
    const float* bn1b   = (const float*)d_in[21];
    const float* Wc2    = (const float*)d_in[22];  // (64, 128)
    const float* bn2g   = (const float*)d_in[23];
    const float* bn2b   = (const float*)d_in[24];
    const float* Wc3    = (const float*)d_in[25];  // (128, 128)
    const float* bn3g   = (const float*)d_in[26];
    const float* bn3b   = (const float*)d_in[27];
    const float* Wc4    = (const float*)d_in[28];  // (256, 256)
    const float* bn4g   = (const float*)d_in[29];
    const float* bn4b   = (const float*)d_in[30];
    const float* Wc5    = (const float*)d_in[31];  // (1024, 512)
    const float* bn5g   = (const float*)d_in[32];
    const float* bn5b   = (const float*)d_in[33];

    // ---- workspace carve-out (floats) ----
    float* wsf = (float*)d_ws;
    size_t cur = 0;
    auto carve = [&](size_t n) -> float* {
        float* p = wsf + cur;
        cur += (n + 63) & ~(size_t)63;
        return p;
    };
    float* xt    = carve((size_t)BATCH * 4 * NPTS);   // zero-padded 4-channel points
    float* xt2   = carve((size_t)BATCH * 4 * NPTS);   // transformed, 4-channel
    float* nrm   = carve((size_t)BATCH * NPTS);
    int*   idx   = (int*)carve((size_t)BATCH * PKN);
    float* tmax  = carve((size_t)BATCH * 128 * NPTS);
    float* gfeat = carve((size_t)BATCH * 1024);
    float* f1    = carve((size_t)BATCH * 512);
    float* f2    = carve((size_t)BATCH * 256);
    float* tmat  = carve((size_t)BATCH * 9);
    float* stats = carve((size_t)2 * 1024);
    float* Wt1p  = carve((size_t)64 * 8);             // padded (64,8) weights
    float* Wc1p  = carve((size_t)64 * 8);
    float* xcat  = carve((size_t)BATCH * 512 * NPTS);
    float* ebuf  = carve((size_t)BATCH * 256 * PKN);  // 41.9M floats, biggest
    float* gram  = ebuf;                              // aliased: disjoint live ranges

    const int XCB = 512 * NPTS;   // xcat batch stride
    const int XTB = 4 * NPTS;     // padded point-cloud batch stride

    // ---- stage 0: transpose + pad input, repack 6-col weights to 8 cols ----
    k_transpose_in4<<<dim3((BATCH * 4 * NPTS + 255) / 256), 256, 0, stream>>>(x_in, xt);
    k_pad_w6to8<<<dim3(2), 256, 0, stream>>>(Wt1, Wt1p);
    k_pad_w6to8<<<dim3(2), 256, 0, stream>>>(Wc1, Wc1p);

    // ---- transform net ----
    run_knn<4>(xt, XTB, nrm, gram, idx, stream);
    float* e1 = ebuf;                                   // (B,64,PKN)
    float* e2 = ebuf + (size_t)BATCH * 64 * PKN;        // (B,128,PKN)
    k_edgeconv_wmma<4><<<dim3(PKN / 256, 64 / 16, BATCH), 128, 0, stream>>>(
        xt, XTB, idx, Wt1p, 64, e1);
    run_bn4d(e1, 64, PKN, tbn1g, tbn1b, stats, stream);
    k_conv_wmma<64><<<dim3(PKN / 256, 128 / 16, BATCH), 128, 0, stream>>>(e1, PKN, Wt2, 128, e2);
    run_bn4d(e2, 128, PKN, tbn2g, tbn2b, stats, stream);
    k_maxpool_k<<<dim3((BATCH * 128 * NPTS + 255) / 256), 256, 0, stream>>>(
        e2, 128, tmax, 128 * NPTS);
    k_conv_wmma<128><<<dim3(NPTS / 256, 1024 / 16, BATCH), 128, 0, stream>>>(
        tmax, NPTS, Wt3, 1024, ebuf);
    run_bn4d(ebuf, 1024, NPTS, tbn3g, tbn3b, stats, stream);
    k_maxpool_n<<<dim3((BATCH * 1024 + 255) / 256), 256, 0, stream>>>(ebuf, 1024, gfeat);
    k_fc<<<dim3((BATCH * 512 + 255) / 256), 256, 0, stream>>>(gfeat, 1024, Wf1, nullptr, 512, f1);
    k_bn_vec_lrelu<<<dim3(2), 256, 0, stream>>>(f1, 512, bnf1g, bnf1b);
    k_fc<<<dim3((BATCH * 256 + 255) / 256), 256, 0, stream>>>(f1, 512, Wf2, bf2, 256, f2);
    k_bn_vec_lrelu<<<dim3(1), 256, 0, stream>>>(f2, 256, bnf2g, bnf2b);
    k_fc3_eye<<<dim3(1), 128, 0, stream>>>(f2, Wf3, bf3, tmat);
    k_apply_transform<<<dim3((BATCH * 4 * NPTS + 255) / 256), 256, 0, stream>>>(xt, tmat, xt2);

    // ---- edge convs (pooled outputs packed into xcat as (B,512,N)) ----
    run_edgeconv<4>(xt2, XTB, Wc1p, 64, bn1g, bn1b, xcat + 0 * NPTS, XCB,
                    nrm, gram, idx, ebuf, stats, stream);
    run_edgeconv<64>(xcat + 0 * NPTS, XCB, Wc2, 64, bn2g, bn2b, xcat + 64 * NPTS, XCB,
                     nrm, gram, idx, ebuf, stats, stream);
    run_edgeconv<64>(xcat + 64 * NPTS, XCB, Wc3, 128, bn3g, bn3b, xcat + 128 * NPTS, XCB,
                     nrm, gram, idx, ebuf, stats, stream);
    run_edgeconv<128>(xcat + 128 * NPTS, XCB, Wc4, 256, bn4g, bn4b, xcat + 256 * NPTS, XCB,
                      nrm, gram, idx, ebuf, stats, stream);

    // ---- conv5 + global max pool ----
    k_conv_wmma<512><<<dim3(NPTS / 256, 1024 / 16, BATCH), 128, 0, stream>>>(
        xcat, NPTS, Wc5, 1024, ebuf);
    run_bn4d(ebuf, 1024, NPTS, bn5g, bn5b, stats, stream);
    k_maxpool_n<<<dim3((BATCH * 1024 + 255) / 256), 256, 0, stream>>>(ebuf, 1024, (float*)d_out);
}